// TimeVaryingLPBiquad_29953101923082
// MI455X (gfx1250) — compile-verified
//
#include <hip/hip_runtime.h>
#include <stdint.h>

// ---------------- problem / tiling constants ----------------
#define TPB     256                    // 8 wave32 per workgroup
#define KSAMP   16                     // samples serially composed per thread
#define TILE    (TPB * KSAMP)          // 4096 samples per workgroup tile
#define BROWS   32
#define TLEN    131072
#define NTILES  (TLEN / TILE)          // 32 tiles per row
#define NBLK    (BROWS * NTILES)       // 1024 workgroups in pass 1/3
#define NCHUNK  (NBLK * TPB)           // 262144 thread-chunks
#define STAGE_IT (TILE / (TPB * 4))    // 4 b128 iterations per array

// log-space modulation constants (match reference)
#define LOG_MIN_W  (-5.8604363f)       // ln(0.00285)
#define DLW        ( 7.0051662f)       // ln(pi) - ln(0.00285)
#define LOG_MIN_Q  (-0.3465832f)       // ln(0.7071)
#define DLQ        ( 1.7328776f)       // ln(4) - ln(0.7071)
#define SFACT      ( 0.999f)           // 1 - EPS

// ---------------- CDNA5 async global<->LDS helpers ----------------
// Generic pointers into LDS carry the LDS byte offset in their low 32 bits
// (aperture check uses ADDR[63:32]; low 32 bits form the LDS address — ISA 10.2).
__device__ __forceinline__ uint32_t lds_off32(const void* p) {
  return (uint32_t)(uintptr_t)p;
}
__device__ __forceinline__ void async_g2l_b128(uint32_t lds, const float* g) {
  asm volatile("global_load_async_to_lds_b128 %0, %1, off"
               :: "v"(lds), "v"(g) : "memory");
}
__device__ __forceinline__ void async_l2g_b128(float* g, uint32_t lds) {
  asm volatile("global_store_async_from_lds_b128 %0, %1, off"
               :: "v"(g), "v"(lds) : "memory");
}
__device__ __forceinline__ void wait_async0() {
  asm volatile("s_wait_asynccnt 0x0" ::: "memory");
}
__device__ __forceinline__ void wait_tensor0() {
#if __has_builtin(__builtin_amdgcn_s_wait_tensorcnt)
  __builtin_amdgcn_s_wait_tensorcnt(0);
  asm volatile("" ::: "memory");
#else
  asm volatile("s_wait_tensorcnt 0x0" ::: "memory");
#endif
}

// ---------------- Tensor Data Mover (TDM) 1D tile load ----------------
#if __has_builtin(__builtin_amdgcn_tensor_load_to_lds)
#define HAVE_TDM 1
typedef unsigned int v4u __attribute__((ext_vector_type(4)));
typedef int          v8i __attribute__((ext_vector_type(8)));
typedef int          v4i __attribute__((ext_vector_type(4)));

// Issue one TDM DMA: n contiguous 4-byte elements, global -> LDS.
// D# layout per CDNA5 ISA.
__device__ __forceinline__ void tdm_load_1d(uint32_t lds_byte_off,
                                            const void* gptr, uint32_t n) {
  const uint64_t ga = (uint64_t)(uintptr_t)gptr;
  v4u g0;
  g0.x = 1u;                                   // count=1 (valid), user mode
  g0.y = lds_byte_off;                         // lds_addr [63:32]
  g0.z = (uint32_t)ga;                         // global_addr[31:0]
  g0.w = (uint32_t)(ga >> 32) | (2u << 30);    // global_addr[56:32] | type=2
  v8i g1;
  g1[0] = (int)(2u << 16);                     // wg_mask=0, data_size=2 (4B)
  g1[1] = (int)((n & 0xFFFFu) << 16);          // atomic_addr=0 | tensor_dim0 lo
  g1[2] = (int)((n >> 16) | (1u << 16));       // tensor_dim0 hi | tensor_dim1=1
  g1[3] = (int)((n & 0xFFFFu) << 16);          // tensor_dim1 hi=0 | tile_dim0=n
  g1[4] = 1;                                   // tile_dim1=1, tile_dim2=0
  g1[5] = (int)n;                              // tensor_dim0_stride[31:0]
  g1[6] = 0;                                   // stride0 hi | stride1 lo
  g1[7] = 0;                                   // stride1 hi
  v4i g2 = {};                                 // <=2D: groups 2/3 zero
  v4i g3 = {};
#if __clang_major__ >= 23
  v8i g4 = {};
  __builtin_amdgcn_tensor_load_to_lds(g0, g1, g2, g3, g4, 0);
#else
  __builtin_amdgcn_tensor_load_to_lds(g0, g1, g2, g3, 0);
#endif
}
#else
#define HAVE_TDM 0
#endif

// ---------------- biquad coefficient math ----------------
__device__ __forceinline__ void biquad_coeff(float wmv, float qmv,
                                             float& a1, float& a2, float& p) {
  float w  = __expf(fmaf(wmv, DLW, LOG_MIN_W));     // v_exp_f32
  float rq = __expf(-fmaf(qmv, DLQ, LOG_MIN_Q));    // 1/q
  float sw, cw;
  __sincosf(w, &sw, &cw);                           // v_sin/v_cos
  float alpha = 0.5f * sw * rq;                     // sin(w)/(2q)
  float ra0   = 1.0f / (1.0f + alpha);
  a1 = (-2.0f * SFACT) * cw * ra0;
  a2 = (SFACT * (1.0f - alpha)) * ra0;
  p  = 0.5f * (1.0f - cw) * ra0;                    // b0=b2=p, b1=2p
}

// =============================================================
// Pass 1: per-thread chunk affine maps + intra-block scan.
// State s[t] = (y[t], y[t-1]);  s[t] = A[t] s[t-1] + (x[t],0),
// A = [[-a1,-a2],[1,0]].  Map = (M 2x2, v 2).
// Tile staging via TDM (wave-level DMA) when available.
// =============================================================
__global__ __launch_bounds__(TPB) void lp_pass1_chunk_scan(
    const float* __restrict__ x, const float* __restrict__ wm,
    const float* __restrict__ qm,
    float* __restrict__ wsPrefix,   // [NCHUNK][6] exclusive per-thread prefix
    float* __restrict__ wsBlock)    // [NBLK][6]  per-workgroup total map
{
  __shared__ __align__(16) float ldsW[TILE];
  __shared__ __align__(16) float ldsQ[TILE];
  __shared__ __align__(16) float ldsX[TILE];
  __shared__ float sc[6][TPB];

  const int tile = blockIdx.x, row = blockIdx.y, tid = threadIdx.x;
  const int base = row * TLEN + tile * TILE;

#if HAVE_TDM
  // TDM ignores EXEC: only wave 0 may reach the issue point (wave-uniform
  // branch), otherwise every wave would launch a duplicate DMA.
  if (tid < 32) {
    tdm_load_1d(lds_off32(ldsW), wm + base, TILE);
    tdm_load_1d(lds_off32(ldsQ), qm + base, TILE);
    tdm_load_1d(lds_off32(ldsX), x  + base, TILE);
    wait_tensor0();            // wave 0's TENSORcnt -> 0: tiles are in LDS
  }
#else
#pragma unroll
  for (int it = 0; it < STAGE_IT; ++it) {
    const int i = tid * 4 + it * (TPB * 4);
    async_g2l_b128(lds_off32(&ldsW[i]), wm + base + i);
    async_g2l_b128(lds_off32(&ldsQ[i]), qm + base + i);
    async_g2l_b128(lds_off32(&ldsX[i]), x  + base + i);
  }
  wait_async0();
#endif
  __syncthreads();             // everyone sees the staged tile

  // ---- serial composition of KSAMP samples (M <- A M, v <- A v + b) ----
  float m00 = 1.f, m01 = 0.f, m10 = 0.f, m11 = 1.f, v0 = 0.f, v1 = 0.f;
  const int l = tid * KSAMP;
#pragma unroll
  for (int j = 0; j < KSAMP; ++j) {
    float a1, a2, p;
    biquad_coeff(ldsW[l + j], ldsQ[l + j], a1, a2, p);
    (void)p;
    float xj  = ldsX[l + j];
    float n00 = -a1 * m00 - a2 * m10;
    float n01 = -a1 * m01 - a2 * m11;
    m10 = m00; m11 = m01; m00 = n00; m01 = n01;
    float nv0 = fmaf(-a1, v0, fmaf(-a2, v1, xj));
    v1 = v0; v0 = nv0;
  }

  // ---- Hillis-Steele inclusive scan of affine maps over the block ----
  sc[0][tid] = m00; sc[1][tid] = m01; sc[2][tid] = m10;
  sc[3][tid] = m11; sc[4][tid] = v0;  sc[5][tid] = v1;
  __syncthreads();
  for (int d = 1; d < TPB; d <<= 1) {
    float e00 = 0.f, e01 = 0.f, e10 = 0.f, e11 = 0.f, ev0 = 0.f, ev1 = 0.f;
    const bool act = (tid >= d);
    if (act) {
      e00 = sc[0][tid - d]; e01 = sc[1][tid - d]; e10 = sc[2][tid - d];
      e11 = sc[3][tid - d]; ev0 = sc[4][tid - d]; ev1 = sc[5][tid - d];
    }
    __syncthreads();
    if (act) {   // later ∘ earlier:  M = Mcur*Me,  v = Mcur*ve + vcur
      float n00 = m00 * e00 + m01 * e10;
      float n01 = m00 * e01 + m01 * e11;
      float n10 = m10 * e00 + m11 * e10;
      float n11 = m10 * e01 + m11 * e11;
      float nv0 = m00 * ev0 + m01 * ev1 + v0;
      float nv1 = m10 * ev0 + m11 * ev1 + v1;
      m00 = n00; m01 = n01; m10 = n10; m11 = n11; v0 = nv0; v1 = nv1;
      sc[0][tid] = m00; sc[1][tid] = m01; sc[2][tid] = m10;
      sc[3][tid] = m11; sc[4][tid] = v0;  sc[5][tid] = v1;
    }
    __syncthreads();
  }

  // exclusive prefix for this thread = inclusive of (tid-1); identity for 0
  float E00 = 1.f, E01 = 0.f, E10 = 0.f, E11 = 0.f, Ev0 = 0.f, Ev1 = 0.f;
  if (tid > 0) {
    E00 = sc[0][tid - 1]; E01 = sc[1][tid - 1]; E10 = sc[2][tid - 1];
    E11 = sc[3][tid - 1]; Ev0 = sc[4][tid - 1]; Ev1 = sc[5][tid - 1];
  }
  const int    bidx = row * NTILES + tile;
  const size_t cidx = (size_t)bidx * TPB + tid;
  float* pp = wsPrefix + 6 * cidx;
  pp[0] = E00; pp[1] = E01; pp[2] = E10; pp[3] = E11; pp[4] = Ev0; pp[5] = Ev1;
  if (tid == TPB - 1) {   // block total (my registers hold the full inclusive)
    float* pb = wsBlock + 6 * (size_t)bidx;
    pb[0] = m00; pb[1] = m01; pb[2] = m10; pb[3] = m11; pb[4] = v0; pb[5] = v1;
  }
}

// =============================================================
// Pass 2: one wave32, lane = batch row; sequential 2x2 matvec
// over the NTILES per-block maps -> state entering each tile.
// =============================================================
__global__ __launch_bounds__(32) void lp_pass2_tile_states(
    const float* __restrict__ wsBlock, float* __restrict__ wsState)
{
  const int r = threadIdx.x;
  if (r >= BROWS) return;
  float s0 = 0.f, s1 = 0.f;                // row starts from zero state
  for (int t = 0; t < NTILES; ++t) {
    const int idx = r * NTILES + t;
    wsState[2 * idx]     = s0;             // state entering tile t
    wsState[2 * idx + 1] = s1;
    const float* pb = wsBlock + 6 * (size_t)idx;
    float n0 = pb[0] * s0 + pb[1] * s1 + pb[4];
    float n1 = pb[2] * s0 + pb[3] * s1 + pb[5];
    s0 = n0; s1 = n1;
  }
}

// =============================================================
// Pass 3: seed each thread chunk via (prefix map)·(tile state),
// regenerate y serially, fuse FIR: out = p*(y0 + 2*y1 + y2),
// write tile back through LDS with async stores.
// =============================================================
__global__ __launch_bounds__(TPB) void lp_pass3_apply(
    const float* __restrict__ x, const float* __restrict__ wm,
    const float* __restrict__ qm,
    const float* __restrict__ wsPrefix, const float* __restrict__ wsState,
    float* __restrict__ out)
{
  __shared__ __align__(16) float ldsW[TILE];
  __shared__ __align__(16) float ldsQ[TILE];
  __shared__ __align__(16) float ldsX[TILE];

  const int tile = blockIdx.x, row = blockIdx.y, tid = threadIdx.x;
  const int base = row * TLEN + tile * TILE;

#pragma unroll
  for (int it = 0; it < STAGE_IT; ++it) {
    const int i = tid * 4 + it * (TPB * 4);
    async_g2l_b128(lds_off32(&ldsW[i]), wm + base + i);
    async_g2l_b128(lds_off32(&ldsQ[i]), qm + base + i);
    async_g2l_b128(lds_off32(&ldsX[i]), x  + base + i);
  }

  const int    bidx = row * NTILES + tile;
  const size_t cidx = (size_t)bidx * TPB + tid;
  const float  sb0  = wsState[2 * bidx];
  const float  sb1  = wsState[2 * bidx + 1];
  const float* pp   = wsPrefix + 6 * cidx;
  const float  E00 = pp[0], E01 = pp[1], E10 = pp[2],
               E11 = pp[3], Ev0 = pp[4], Ev1 = pp[5];
  // state entering my chunk: (y[t0-1], y[t0-2])
  float y1 = E00 * sb0 + E01 * sb1 + Ev0;
  float y2 = E10 * sb0 + E11 * sb1 + Ev1;

  wait_async0();
  __syncthreads();

  const int l = tid * KSAMP;
#pragma unroll
  for (int j = 0; j < KSAMP; ++j) {
    float a1, a2, p;
    biquad_coeff(ldsW[l + j], ldsQ[l + j], a1, a2, p);
    float xj = ldsX[l + j];
    float y0 = fmaf(-a1, y1, fmaf(-a2, y2, xj));        // IIR step
    ldsX[l + j] = p * (y0 + 2.0f * y1 + y2);            // fused FIR
    y2 = y1; y1 = y0;
  }
  __syncthreads();   // all output in LDS (also drains DScnt before async read)

#pragma unroll
  for (int it = 0; it < STAGE_IT; ++it) {
    const int i = tid * 4 + it * (TPB * 4);
    async_l2g_b128(out + base + i, lds_off32(&ldsX[i]));
  }
  wait_async0();
}

// =============================================================
extern "C" void kernel_launch(void* const* d_in, const int* in_sizes, int n_in,
                              void* d_out, int out_size, void* d_ws, size_t ws_size,
                              hipStream_t stream) {
  (void)in_sizes; (void)n_in; (void)out_size; (void)ws_size;
  const float* x  = (const float*)d_in[0];
  const float* wm = (const float*)d_in[1];
  const float* qm = (const float*)d_in[2];
  // d_in[3] = interp_coeff (unused by the reference math)
  float* out      = (float*)d_out;

  float* wsPrefix = (float*)d_ws;                         // NCHUNK*6 floats
  float* wsBlock  = wsPrefix + (size_t)6 * NCHUNK;        // NBLK*6
  float* wsState  = wsBlock  + (size_t)6 * NBLK;          // NBLK*2

  dim3 grid(NTILES, BROWS), blk(TPB);
  lp_pass1_chunk_scan<<<grid, blk, 0, stream>>>(x, wm, qm, wsPrefix, wsBlock);
  lp_pass2_tile_states<<<1, 32, 0, stream>>>(wsBlock, wsState);
  lp_pass3_apply<<<grid, blk, 0, stream>>>(x, wm, qm, wsPrefix, wsState, out);
}